// GDAttention_11373073399827
// MI455X (gfx1250) — compile-verified
//
#include <hip/hip_runtime.h>
#include <math.h>
#include <stdint.h>

#define NH   8
#define BB   4
#define SS   4096
#define DD   128
#define QBLK 64
#define KBLK 16
#define NWAVE 4
#define PADD 132   // padded LDS row stride (floats) to break bank conflicts

typedef __attribute__((ext_vector_type(2))) float v2f;
typedef __attribute__((ext_vector_type(8))) float v8f;
typedef __attribute__((ext_vector_type(4))) float f4;

__device__ __forceinline__ v8f wmma4(v2f a, v2f b, v8f c) {
  // D = A(16x4,f32) * B(4x16,f32) + C(16x16,f32)
  return __builtin_amdgcn_wmma_f32_16x16x4_f32(false, a, false, b, (short)0, c,
                                               false, false);
}

// Async global->LDS copy, 16B per lane, tracked by ASYNCcnt.
// LDS aperture: addr[31:0] of the flat shared pointer is the LDS byte offset.
__device__ __forceinline__ void async_b128(void* lds_dst, const void* gsrc) {
  unsigned ldsoff = (unsigned)(uintptr_t)lds_dst;
  asm volatile("global_load_async_to_lds_b128 %0, %1, off"
               :: "v"(ldsoff), "v"(gsrc)
               : "memory");
}
#define S_WAIT_ASYNC(N) asm volatile("s_wait_asynccnt " #N ::: "memory")

__global__ __launch_bounds__(128) void gd_attn_kernel(
    const float* __restrict__ e, const float* __restrict__ p,
    const float* __restrict__ W, const float* __restrict__ bias,
    float* __restrict__ out) {
  __shared__ __align__(16) float Ws[DD][PADD];    // head-folded W_o (128x128)
  __shared__ __align__(16) float Qs[QBLK][PADD];  // Q tile; reused for ctx
  __shared__ __align__(16) float Ks[2][KBLK][PADD];  // double-buffered K tile
  __shared__ __align__(16) float Es[2][KBLK][PADD];  // double-buffered V tile
  __shared__ float Ps[NWAVE][16][20];             // score/prob tiles per wave
  __shared__ float alphaS[NWAVE][16];
  __shared__ float mS[NWAVE][16];
  __shared__ float lS[NWAVE][16];

  const int tid  = threadIdx.x;
  const int wave = tid >> 5;
  const int lane = tid & 31;
  const int lm   = lane & 15;   // M (A/C row) or N (B/C col) within 16
  const int hi   = lane >> 4;   // half-wave select
  const int koff = hi * 2;      // A/B layout: lanes 0-15 hold K={0,1}, 16-31 K={2,3}

  const int bid    = blockIdx.x;
  const int b      = bid / (SS / QBLK);
  const int qblock = (bid % (SS / QBLK)) * QBLK;
  const int qw     = qblock + wave * 16;  // this wave's first query row

  const float* pb = p + (size_t)b * (SS + 1) * DD;
  const float* eb = e + (size_t)b * SS * DD;

  const int nkt = qblock / 16 + NWAVE;  // causal K-tile count for this block

  // Kick off async load of K/V tile 0 (8 x b128 per wave -> ASYNCcnt=8)
  {
    const float* ksrc = pb;  // k = p[:, :-1, :]
    const float* esrc = eb;
    for (int idx = tid; idx < KBLK * 32; idx += 128) {
      int r = idx >> 5, c4 = idx & 31;
      async_b128(&Ks[0][r][c4 * 4], ksrc + (size_t)r * DD + c4 * 4);
      async_b128(&Es[0][r][c4 * 4], esrc + (size_t)r * DD + c4 * 4);
    }
  }

  // Fold heads: Ws[d][dp] = sum_h W[d][h*128+dp]  (tile(ctx,8)@W.T == ctx@Ws.T)
  for (int idx = tid; idx < DD * DD; idx += 128) {
    int d = idx >> 7, dp = idx & 127;
    const float* wrow = W + (size_t)d * (DD * NH) + dp;
    float s = 0.f;
#pragma unroll
    for (int h = 0; h < NH; ++h) s += wrow[h * DD];
    Ws[d][dp] = s;
  }

  // Q tile: q[r] = p[b, qblock+r+1, :]
  for (int idx = tid; idx < QBLK * (DD / 4); idx += 128) {
    int r = idx >> 5, c4 = idx & 31;
    f4 v = *(const f4*)(pb + (size_t)(qblock + r + 1) * DD + c4 * 4);
    *(f4*)&Qs[r][c4 * 4] = v;
  }

  if (lane < 16) { mS[wave][lane] = -INFINITY; lS[wave][lane] = 0.f; }
  __syncthreads();

  // Hoist this wave's Q stripe into registers, A-matrix layout (64 VGPRs):
  // reused across every K tile, halves inner-loop LDS traffic.
  v2f qa[32];
#pragma unroll
  for (int kc = 0; kc < 32; ++kc) {
    qa[kc].x = Qs[wave * 16 + lm][kc * 4 + koff];
    qa[kc].y = Qs[wave * 16 + lm][kc * 4 + koff + 1];
  }

  v8f acc[8];  // ctx accumulator: 16 rows x 128 cols (8 N-tiles of C)
#pragma unroll
  for (int t = 0; t < 8; ++t) acc[t] = {};

  for (int kt = 0; kt < nkt; ++kt) {
    const int buf = kt & 1;

    // Issue async load of the NEXT tile into the other buffer (its previous
    // consumer finished at the end-of-iteration barrier of kt-1).
    if (kt + 1 < nkt) {
      const float* ksrc = pb + (size_t)(kt + 1) * KBLK * DD;
      const float* esrc = eb + (size_t)(kt + 1) * KBLK * DD;
      const int nb = buf ^ 1;
      for (int idx = tid; idx < KBLK * 32; idx += 128) {
        int r = idx >> 5, c4 = idx & 31;
        async_b128(&Ks[nb][r][c4 * 4], ksrc + (size_t)r * DD + c4 * 4);
        async_b128(&Es[nb][r][c4 * 4], esrc + (size_t)r * DD + c4 * 4);
      }
      S_WAIT_ASYNC(8);  // async loads retire in order: tile kt is now in LDS
    } else {
      S_WAIT_ASYNC(0);
    }
    __syncthreads();  // tile kt visible to all waves

    // S = Q * Ktile^T  (reduce D=128 in 32 steps of K=4); A from registers
    v8f sc = {};
#pragma unroll
    for (int kc = 0; kc < 32; ++kc) {
      v2f bm;
      bm.x = Ks[buf][lm][kc * 4 + koff];
      bm.y = Ks[buf][lm][kc * 4 + koff + 1];
      sc = wmma4(qa[kc], bm, sc);
    }
#pragma unroll
    for (int v = 0; v < 8; ++v) Ps[wave][v + hi * 8][lm] = sc[v];
    __syncthreads();

    // Online softmax, one row per lane (lanes 0-15), causal mask via nvalid
    if (lane < 16) {
      const int qg = qw + lane;
      int nvalid = qg - kt * KBLK + 1;
      nvalid = nvalid > 16 ? 16 : nvalid;
      if (nvalid <= 0) {
        alphaS[wave][lane] = 1.f;
#pragma unroll
        for (int j = 0; j < 16; ++j) Ps[wave][lane][j] = 0.f;
      } else {
        const float scale = 0.088388347648318447f;  // 1/sqrt(128)
        float sv[16], mx = -INFINITY;
#pragma unroll
        for (int j = 0; j < 16; ++j) {
          sv[j] = Ps[wave][lane][j] * scale;
          if (j < nvalid && sv[j] > mx) mx = sv[j];
        }
        float mold  = mS[wave][lane];
        float mnew  = mold > mx ? mold : mx;
        float alpha = __expf(mold - mnew);
        float lsum  = 0.f;
#pragma unroll
        for (int j = 0; j < 16; ++j) {
          float pv = (j < nvalid) ? __expf(sv[j] - mnew) : 0.f;
          Ps[wave][lane][j] = pv;
          lsum += pv;
        }
        mS[wave][lane]     = mnew;
        lS[wave][lane]     = lS[wave][lane] * alpha + lsum;
        alphaS[wave][lane] = alpha;
      }
    }
    __syncthreads();

    // Rescale accumulator, then ctx += P * Etile
    float al[8];
#pragma unroll
    for (int v = 0; v < 8; ++v) al[v] = alphaS[wave][v + hi * 8];
#pragma unroll
    for (int t = 0; t < 8; ++t)
#pragma unroll
      for (int v = 0; v < 8; ++v) acc[t][v] *= al[v];

#pragma unroll
    for (int kc = 0; kc < 4; ++kc) {
      v2f a;
      a.x = Ps[wave][lm][kc * 4 + koff];
      a.y = Ps[wave][lm][kc * 4 + koff + 1];
#pragma unroll
      for (int t = 0; t < 8; ++t) {
        v2f bm;
        bm.x = Es[buf][kc * 4 + koff][t * 16 + lm];
        bm.y = Es[buf][kc * 4 + koff + 1][t * 16 + lm];
        acc[t] = wmma4(a, bm, acc[t]);
      }
    }
    __syncthreads();  // all waves done with buf before it is async-overwritten
  }

  // Normalize ctx and stage to LDS (Qs is dead now)
  float rl[8];
#pragma unroll
  for (int v = 0; v < 8; ++v) rl[v] = 1.f / lS[wave][v + hi * 8];
#pragma unroll
  for (int t = 0; t < 8; ++t)
#pragma unroll
    for (int v = 0; v < 8; ++v)
      Qs[wave * 16 + v + hi * 8][t * 16 + lm] = acc[t][v] * rl[v];
  __syncthreads();

  // y = ctx @ Ws^T  (per-wave 16x128 output)
#pragma unroll
  for (int t = 0; t < 8; ++t) acc[t] = {};
  for (int kc = 0; kc < 32; ++kc) {
    v2f a;
    a.x = Qs[wave * 16 + lm][kc * 4 + koff];
    a.y = Qs[wave * 16 + lm][kc * 4 + koff + 1];
#pragma unroll
    for (int t = 0; t < 8; ++t) {
      v2f bm;
      bm.x = Ws[t * 16 + lm][kc * 4 + koff];
      bm.y = Ws[t * 16 + lm][kc * 4 + koff + 1];
      acc[t] = wmma4(a, bm, acc[t]);
    }
  }

  // Bias + store (C layout: col = t*16+lm, row = qw + v + hi*8)
#pragma unroll
  for (int t = 0; t < 8; ++t) {
    const int n  = t * 16 + lm;
    const float bn = bias[n];
#pragma unroll
    for (int v = 0; v < 8; ++v) {
      const int row = qw + v + hi * 8;
      out[((size_t)b * SS + row) * DD + n] = acc[t][v] + bn;
    }
  }
}

extern "C" void kernel_launch(void* const* d_in, const int* in_sizes, int n_in,
                              void* d_out, int out_size, void* d_ws,
                              size_t ws_size, hipStream_t stream) {
  const float* e    = (const float*)d_in[0];
  const float* p    = (const float*)d_in[1];
  const float* W    = (const float*)d_in[2];
  const float* bias = (const float*)d_in[3];
  float* out        = (float*)d_out;

  dim3 grid(BB * (SS / QBLK));  // 256 workgroups
  dim3 block(128);              // 4 wave32s
  hipLaunchKernelGGL(gd_attn_kernel, grid, block, 0, stream, e, p, W, bias, out);
}